// BayesClassifier_2980707304087
// MI455X (gfx1250) — compile-verified
//
#include <hip/hip_runtime.h>
#include <hip/hip_bf16.h>
#include <math.h>

// Problem constants (reference: B=32768, C=100, D=64)
#define GMM_B 32768
#define GMM_C 100
#define GMM_D 64

typedef __attribute__((ext_vector_type(2))) float v2f;
typedef __attribute__((ext_vector_type(8))) float v8f;

// ---------------------------------------------------------------------------
// Kernel 1: per-class prep. One block (64 threads) per class.
//   - Cholesky of cov (in LDS)
//   - G = L^-1 (forward substitution of identity, thread-per-column)
//   - h = G * mu
//   - cc = log w - 0.5*(logdet + D*log(2*pi))
// ---------------------------------------------------------------------------
__global__ __launch_bounds__(64)
void gmm_prep_kernel(const float* __restrict__ means,
                     const float* __restrict__ covs,
                     const float* __restrict__ weights,
                     float* __restrict__ Gout,
                     float* __restrict__ hout,
                     float* __restrict__ ccout)
{
    __shared__ float A [GMM_D * 65];   // cov -> L (lower triangular)
    __shared__ float Gi[GMM_D * 65];   // L^-1
    __shared__ float red[GMM_D];
    __shared__ float mu [GMM_D];

    const int t = threadIdx.x;     // 0..63, owns row t (Cholesky) / column t (inverse)
    const int c = blockIdx.x;      // class

    // Load cov row t and mean
    for (int k = 0; k < GMM_D; ++k)
        A[t * 65 + k] = covs[(size_t)c * GMM_D * GMM_D + (size_t)t * GMM_D + k];
    mu[t] = means[c * GMM_D + t];
    __syncthreads();

    // Right-looking Cholesky: after step j, column j of L is final.
    for (int j = 0; j < GMM_D; ++j) {
        if (t == j) A[j * 65 + j] = sqrtf(A[j * 65 + j]);
        __syncthreads();
        if (t > j) A[t * 65 + j] /= A[j * 65 + j];
        __syncthreads();
        if (t > j) {
            const float ltj = A[t * 65 + j];
            for (int k = j + 1; k <= t; ++k)
                A[t * 65 + k] -= ltj * A[k * 65 + j];
        }
        __syncthreads();
    }

    // logdet = 2 * sum(log(diag(L)))
    red[t] = 2.0f * logf(A[t * 65 + t]);
    __syncthreads();
    if (t == 0) {
        float ld = 0.0f;
        for (int k = 0; k < GMM_D; ++k) ld += red[k];
        const float LOG2PI = 1.8378770664093453f;
        ccout[c] = logf(weights[c]) - 0.5f * (ld + (float)GMM_D * LOG2PI);
    }

    // G = L^-1, thread t computes column t: L g = e_t (no cross-thread deps)
    for (int i = 0; i < GMM_D; ++i) {
        float sum = (i == t) ? 1.0f : 0.0f;
        for (int k = t; k < i; ++k)
            sum -= A[i * 65 + k] * Gi[k * 65 + t];
        Gi[i * 65 + t] = (i >= t) ? (sum / A[i * 65 + i]) : 0.0f;
    }
    __syncthreads();

    // h[t] = (G * mu)[t]  (row t of G is nonzero only for k <= t)
    float hv = 0.0f;
    for (int k = 0; k <= t; ++k) hv += Gi[t * 65 + k] * mu[k];
    hout[c * GMM_D + t] = hv;

    // Write G row-major [C][D][D]
    for (int k = 0; k < GMM_D; ++k)
        Gout[(size_t)c * GMM_D * GMM_D + (size_t)t * GMM_D + k] = Gi[t * 65 + k];
}

// ---------------------------------------------------------------------------
// Kernel 2: main WMMA kernel.
// Grid: (B/128, C). Block: 256 threads = 8 waves.
// Workgroup computes logits for one class c and 128 batch rows.
// Each wave owns a 16-column N-tile; loops over 4 M-tiles of
// Y = G_c * X^T with v_wmma_f32_16x16x4_f32 (K-loop of 16 steps),
// then reduces sum_m (Y[m,n] - h[m])^2 in-register.
//
// LDS fragment layout (stride 68 floats = 272B: 16B-aligned rows,
// conflict-free for paired-K b64 reads):
//   A frag (16x4 of G): lane l -> G[m0+(l&15)][k0+2*(l>>4) .. +1]
//   B frag (4x16 of X^T): lane l -> x[n0+(l&15)][k0+2*(l>>4) .. +1]
// C/D frag: VGPR j holds row M=j (lanes 0-15) / M=j+8 (lanes 16-31), col N=lane&15.
// ---------------------------------------------------------------------------
#define TILE_N 128
#define LSTR   68

__global__ __launch_bounds__(256)
void gmm_logits_kernel(const float* __restrict__ x,
                       const float* __restrict__ G,
                       const float* __restrict__ h,
                       const float* __restrict__ cc,
                       float* __restrict__ out)
{
    __shared__ __align__(16) float Xl[TILE_N * LSTR];  // X tile: [n][k]
    __shared__ __align__(16) float Gl[GMM_D * LSTR];   // G:      [m][k]
    __shared__ float hl[GMM_D];

    const int t  = threadIdx.x;
    const int c  = blockIdx.y;
    const int n0 = blockIdx.x * TILE_N;

    // ---- Stage tiles (coalesced float4 global reads) ----
    {
        const int col4 = (t & 15) * 4;   // 0..60
        const int r0   = t >> 4;         // 0..15
        #pragma unroll
        for (int it = 0; it < 8; ++it) { // 128 rows of x
            const int r = r0 + it * 16;
            const float4 v = *(const float4*)(x + (size_t)(n0 + r) * GMM_D + col4);
            *(float4*)&Xl[r * LSTR + col4] = v;
        }
        #pragma unroll
        for (int it = 0; it < 4; ++it) { // 64 rows of G
            const int r = r0 + it * 16;
            const float4 v = *(const float4*)(G + (size_t)c * GMM_D * GMM_D + (size_t)r * GMM_D + col4);
            *(float4*)&Gl[r * LSTR + col4] = v;
        }
        if (t < GMM_D) hl[t] = h[c * GMM_D + t];
    }
    __syncthreads();

    // ---- WMMA compute: full EXEC (uniform control flow) ----
    const int lane  = t & 31;
    const int w     = t >> 5;            // wave id = N-tile id (0..7)
    const int nloc  = lane & 15;
    const int khalf = (lane >> 4) * 2;   // 0 or 2: K sub-pair for this half-wave

    const float* brow = &Xl[(w * 16 + nloc) * LSTR];

    float s = 0.0f;
    #pragma unroll
    for (int mt = 0; mt < 4; ++mt) {
        v8f acc = {};
        const float* arow = &Gl[(mt * 16 + nloc) * LSTR];
        #pragma unroll
        for (int k0 = 0; k0 < GMM_D; k0 += 4) {
            v2f a = *(const v2f*)(arow + k0 + khalf);
            v2f b = *(const v2f*)(brow + k0 + khalf);
            // D = A(16x4) * B(4x16) + C ; fp32 WMMA
            acc = __builtin_amdgcn_wmma_f32_16x16x4_f32(
                      /*neg_a=*/false, a, /*neg_b=*/false, b,
                      /*c_mod=*/(short)0, acc,
                      /*reuse_a=*/false, /*reuse_b=*/false);
        }
        const int mrow = mt * 16 + (lane >> 4) * 8;
        #pragma unroll
        for (int j = 0; j < 8; ++j) {
            const float d = acc[j] - hl[mrow + j];
            s += d * d;
        }
    }

    // Fold the two half-wave row groups: lanes l and l+16 hold the same column n.
    s += __shfl_xor(s, 16, 32);

    if (lane < 16) {
        const int b = n0 + w * 16 + nloc;
        out[(size_t)b * GMM_C + c] = cc[c] - 0.5f * s;
    }
}

// ---------------------------------------------------------------------------
// Launch: inputs in setup_inputs() order: x, means, covs, weights (all fp32).
// Workspace: G [C*D*D] + h [C*D] + cc [C] floats (~1.7 MB).
// ---------------------------------------------------------------------------
extern "C" void kernel_launch(void* const* d_in, const int* in_sizes, int n_in,
                              void* d_out, int out_size, void* d_ws, size_t ws_size,
                              hipStream_t stream) {
    (void)in_sizes; (void)n_in; (void)out_size; (void)ws_size;

    const float* x       = (const float*)d_in[0];
    const float* means   = (const float*)d_in[1];
    const float* covs    = (const float*)d_in[2];
    const float* weights = (const float*)d_in[3];
    float* out = (float*)d_out;

    float* Gw  = (float*)d_ws;
    float* hw  = Gw + (size_t)GMM_C * GMM_D * GMM_D;
    float* ccw = hw + (size_t)GMM_C * GMM_D;

    gmm_prep_kernel<<<GMM_C, 64, 0, stream>>>(means, covs, weights, Gw, hw, ccw);

    dim3 grid(GMM_B / TILE_N, GMM_C);
    gmm_logits_kernel<<<grid, 256, 0, stream>>>(x, Gw, hw, ccw, out);
}